// FSMRELoss_35639638622669
// MI455X (gfx1250) — compile-verified
//
#include <hip/hip_runtime.h>
#include <math.h>

typedef float v2f __attribute__((ext_vector_type(2)));
typedef float v4f __attribute__((ext_vector_type(4)));
typedef float v8f __attribute__((ext_vector_type(8)));

#define NB 128      // N
#define LB 32       // L
#define NPAIRS (8 * 128 * 128)       // B*N*N = 131072
#define WAVES_PER_BLOCK 32
#define BLOCKS1 (NPAIRS / WAVES_PER_BLOCK)   // 4096

__device__ __forceinline__ float bperm_f(float v, int srcLane) {
    return __int_as_float(__builtin_amdgcn_ds_bpermute(srcLane << 2, __float_as_int(v)));
}

// Kernel 1: one wave per (b,i,j) pair; lane k handles label-row k (32 floats).
// Emits per-block partials: ce_sum, correct_count, multi_count.
__global__ __launch_bounds__(1024) void fsmre_pair_kernel(
    const float* __restrict__ inp, const int* __restrict__ tgt,
    float* __restrict__ ws_ce, float* __restrict__ ws_corr, float* __restrict__ ws_multi)
{
    __shared__ float lds_ce[WAVES_PER_BLOCK];
    __shared__ float lds_corr[WAVES_PER_BLOCK];
    __shared__ float lds_multi[WAVES_PER_BLOCK];

    const int tid  = threadIdx.x;
    const int lane = tid & 31;
    const int wave = tid >> 5;
    const int pair = blockIdx.x * WAVES_PER_BLOCK + wave;   // b*N*N + i*N + j
    const int j = pair & (NB - 1);
    const int i = (pair >> 7) & (NB - 1);
    const bool offd = (i != j);             // wave-uniform

    float ce = 0.0f;
    bool correct = true;

    if (offd) {
        // Row of lane k: inp + pair*L*L + k*L   (128 bytes, 16B-aligned)
        const v4f* row = (const v4f*)(inp + (((size_t)pair) << 10) + ((size_t)lane << 5));
        const int c_sel = lane >> 2;
        const int e_sel = lane & 3;
        float m = 0.0f;                     // inputs are in (0,1): 0 is a safe -inf
        float p = 0.0f;
#pragma unroll
        for (int c = 0; c < 8; ++c) {
            v4f v = __builtin_nontemporal_load(row + c);
            m = fmaxf(m, fmaxf(fmaxf(v.x, v.y), fmaxf(v.z, v.w)));
            if (c == c_sel) {               // pick diag element p = row[lane]
                float lo = (e_sel & 1) ? v.y : v.x;
                float hi = (e_sel & 1) ? v.w : v.z;
                p = (e_sel & 2) ? hi : lo;
            }
        }
        const int g = __builtin_nontemporal_load(tgt + ((size_t)pair << 5) + lane);
        ce = -logf(g ? p : (1.0f - p));     // -log(p^g * (1-p)^(1-g)), g in {0,1}
        const bool is_max = (p == m);
        correct = g ? is_max : !is_max;
    }

    // per-label / per-pair accuracy via wave32 ballot (L == wave width == 32)
    unsigned bal = (unsigned)__builtin_amdgcn_ballot_w32(correct);
    float corrCnt = offd ? (float)__popc(bal) : 0.0f;
    float multiV  = (offd && bal == 0xFFFFFFFFu) ? 1.0f : 0.0f;

    // exact f32 wave-sum of ce
#if __has_builtin(__builtin_amdgcn_wmma_f32_16x16x4_f32)
    // A(16x4): a.x -> K0 (lanes 0-15) / K2 (lanes 16-31); a.y -> K1/K3 (zero).
    // B(4x16) = ones  =>  D[m][n] = ce[m] + ce[m+16]
    v2f a; a.x = ce;   a.y = 0.0f;
    v2f b; b.x = 1.0f; b.y = 1.0f;
    v8f c = {};
    v8f d = __builtin_amdgcn_wmma_f32_16x16x4_f32(
        false, a, false, b, (short)0, c, false, false);
    float s = ((d[0] + d[1]) + (d[2] + d[3])) + ((d[4] + d[5]) + (d[6] + d[7]));
    float ceW = s + bperm_f(s, lane ^ 16);  // rows 0-7 (this half) + rows 8-15 (other half)
#else
    float ceW = ce;
#pragma unroll
    for (int off = 16; off > 0; off >>= 1) ceW += bperm_f(ceW, lane ^ off);
#endif

    if (lane == 0) {
        lds_ce[wave]    = ceW;
        lds_corr[wave]  = corrCnt;
        lds_multi[wave] = multiV;
    }
    __syncthreads();

    if (tid == 0) {
        float sc = 0.0f, sr = 0.0f, sm = 0.0f;
        for (int w = 0; w < WAVES_PER_BLOCK; ++w) {
            sc += lds_ce[w]; sr += lds_corr[w]; sm += lds_multi[w];
        }
        ws_ce[blockIdx.x]    = sc;
        ws_corr[blockIdx.x]  = sr;
        ws_multi[blockIdx.x] = sm;
    }
}

// Kernel 2: deterministic final reduction of 4096 partials -> 3 scalars.
__global__ __launch_bounds__(256) void fsmre_finalize_kernel(
    const float* __restrict__ ws_ce, const float* __restrict__ ws_corr,
    const float* __restrict__ ws_multi, float* __restrict__ out)
{
    __shared__ float sc[256], sr[256], sm[256];
    const int t = threadIdx.x;
    float a = 0.0f, b = 0.0f, c = 0.0f;
    for (int idx = t; idx < BLOCKS1; idx += 256) {
        a += ws_ce[idx]; b += ws_corr[idx]; c += ws_multi[idx];
    }
    sc[t] = a; sr[t] = b; sm[t] = c;
    __syncthreads();
    if (t == 0) {
        float A = 0.0f, Bv = 0.0f, C = 0.0f;
        for (int k = 0; k < 256; ++k) { A += sc[k]; Bv += sr[k]; C += sm[k]; }
        // num = N*(N-1)*L = 520192 ; loss denom = num * B = 4161536
        out[0] = A  / 4161536.0f;   // loss
        out[1] = Bv / 4161536.0f;   // single_acc: / (B*N*(N-1)*L)
        out[2] = C  / 130048.0f;    // multi_acc:  / (B*N*(N-1))
    }
}

extern "C" void kernel_launch(void* const* d_in, const int* in_sizes, int n_in,
                              void* d_out, int out_size, void* d_ws, size_t ws_size,
                              hipStream_t stream) {
    (void)in_sizes; (void)n_in; (void)out_size; (void)ws_size;
    const float* inp = (const float*)d_in[0];
    const int*   tgt = (const int*)d_in[1];
    float* ws       = (float*)d_ws;
    float* ws_ce    = ws;
    float* ws_corr  = ws + BLOCKS1;
    float* ws_multi = ws + 2 * BLOCKS1;     // 48 KB total scratch

    fsmre_pair_kernel<<<BLOCKS1, 1024, 0, stream>>>(inp, tgt, ws_ce, ws_corr, ws_multi);
    fsmre_finalize_kernel<<<1, 256, 0, stream>>>(ws_ce, ws_corr, ws_multi, (float*)d_out);
}